// ConstraintViolationLoss_23570780520662
// MI455X (gfx1250) — compile-verified
//
#include <hip/hip_runtime.h>

typedef float v2f __attribute__((ext_vector_type(2)));
typedef float v8f __attribute__((ext_vector_type(8)));

#define LAMBDA_MEAN 1.0f
#define LAMBDA_MAX  0.1f

// ---------------- init: zero x, Ax, accumulators ----------------
__global__ void init_ws_kernel(float* __restrict__ x, int n_vars,
                               float* __restrict__ Ax, int n_con,
                               float* __restrict__ accum) {
  int i = blockIdx.x * blockDim.x + threadIdx.x;
  if (i < n_vars) x[i] = 0.0f;
  if (i < n_con)  Ax[i] = 0.0f;
  if (i < 4)      accum[i] = 0.0f;   // [0]=sum, [1]=max(bits), [2]=count
}

// ---------------- generic scatter: x[idx[i]] = vals[i] ----------------
__global__ void scatter_kernel(float* __restrict__ x,
                               const long long* __restrict__ idx,
                               const float* __restrict__ vals, int n) {
  int i = blockIdx.x * blockDim.x + threadIdx.x;
  if (i < n) x[idx[i]] = vals[i];
}

// ---------------- softmax expected-value via V_WMMA_F32_16X16X4_F32 ----------
// One wave per 16-row tile. A = exp(logits - rowmax) as a 16x16 (K) tile fed
// through 4 chained 16x16x4 f32 WMMAs. B col0 = 1 (denominator), col1 = k
// (numerator), other cols 0. expected = D[:,1]/D[:,0] + offset.
__global__ void softmax_ev_wmma_kernel(float* __restrict__ x,
                                       const float* __restrict__ logits,
                                       const float* __restrict__ offs,
                                       const long long* __restrict__ idx,
                                       int ns) {
  const int lane = threadIdx.x & 31;
  const int wave = threadIdx.x >> 5;
  const int tile = blockIdx.x * (blockDim.x >> 5) + wave;
  const int ntiles = (ns + 15) >> 4;
  if (tile >= ntiles) return;                 // wave-uniform exit: EXEC stays all-1s

  const int m    = lane & 15;                 // row within tile
  const int half = lane >> 4;                 // 0: K={4k,4k+1}, 1: K={4k+2,4k+3}
  const int row  = (tile << 4) + m;
  const bool rvalid = row < ns;

  // load this lane's 8 logits (2 per K-block), compute row max across the pair
  // of lanes {m, m+16} which together hold all 16 columns of row m
  v2f a[4];
  float rmax = -3.4e38f;
#pragma unroll
  for (int kb = 0; kb < 4; ++kb) {
    v2f v;
    if (rvalid) v = *(const v2f*)(logits + (long long)row * 16 + kb * 4 + half * 2);
    else        { v.x = 0.0f; v.y = 0.0f; }
    a[kb] = v;
    rmax = fmaxf(rmax, fmaxf(v.x, v.y));
  }
  rmax = fmaxf(rmax, __shfl_xor(rmax, 16, 32));

#pragma unroll
  for (int kb = 0; kb < 4; ++kb) {
    a[kb].x = __expf(a[kb].x - rmax);
    a[kb].y = __expf(a[kb].y - rmax);
  }

  const int n = lane & 15;                    // column of B held by this lane
  v8f c = {};
#pragma unroll
  for (int kb = 0; kb < 4; ++kb) {
    const int k0 = kb * 4 + half * 2;         // global K of this lane's b.x
    v2f b;
    b.x = (n == 0) ? 1.0f : ((n == 1) ? (float)(k0)     : 0.0f);
    b.y = (n == 0) ? 1.0f : ((n == 1) ? (float)(k0 + 1) : 0.0f);
    c = __builtin_amdgcn_wmma_f32_16x16x4_f32(
        /*neg_a=*/false, a[kb], /*neg_b=*/false, b,
        /*c_mod=*/(short)0, c, /*reuse_a=*/false, /*reuse_b=*/false);
  }

  // D layout: VGPR v holds M=v (lanes 0-15, N=lane) and M=v+8 (lanes 16-31).
  // Column 0 lives in lanes {0,16}, column 1 in lanes {1,17}.
  float den = 0.0f, num = 0.0f;
#pragma unroll
  for (int mr = 0; mr < 8; ++mr) {
    float cv   = c[mr];
    float d_lo = __shfl(cv,  0, 32);   // D[mr][0]
    float d_hi = __shfl(cv, 16, 32);   // D[mr+8][0]
    float n_lo = __shfl(cv,  1, 32);   // D[mr][1]
    float n_hi = __shfl(cv, 17, 32);   // D[mr+8][1]
    if (m == mr)     { den = d_lo; num = n_lo; }
    if (m == mr + 8) { den = d_hi; num = n_hi; }
  }

  if (lane < 16 && rvalid) {
    x[idx[row]] = num / den + offs[row];
  }
}

// ---------------- continuous vars take LP solution (applied AFTER scatters) ---
__global__ void apply_lp_kernel(float* __restrict__ x,
                                const float* __restrict__ vf,
                                const long long* __restrict__ vt, int n_vars) {
  int i = blockIdx.x * blockDim.x + threadIdx.x;
  if (i < n_vars && vt[i] == 0) x[i] = vf[(long long)i * 16 + 8];
}

// ---------------- SpMV: gather x along edges, atomic scatter-add per row -----
__global__ void spmv_kernel(const long long* __restrict__ eidx,
                            const float* __restrict__ ef,
                            const float* __restrict__ x,
                            float* __restrict__ Ax, int E) {
  int e = blockIdx.x * blockDim.x + threadIdx.x;
  if (e < E) {
    long long r = eidx[e];
    long long cidx = eidx[(long long)E + e];
    atomicAdd(&Ax[r], ef[e] * x[cidx]);
  }
}

// ---------------- violation reduction: sum, max, count -----------------------
__global__ void reduce_kernel(const float* __restrict__ Ax,
                              const float* __restrict__ cf,
                              float* __restrict__ accum, int n_con) {
  int i = blockIdx.x * blockDim.x + threadIdx.x;
  float v = 0.0f, cnt = 0.0f;
  if (i < n_con) {
    float t = Ax[i] - cf[(long long)i * 8 + 1];
    v = fmaxf(t, 0.0f);
    cnt = (v > 1e-6f) ? 1.0f : 0.0f;
  }
  float s = v, mx = v, cc = cnt;
#pragma unroll
  for (int off = 16; off > 0; off >>= 1) {
    s  += __shfl_down(s, off, 32);
    mx  = fmaxf(mx, __shfl_down(mx, off, 32));
    cc += __shfl_down(cc, off, 32);
  }
  if ((threadIdx.x & 31) == 0) {
    atomicAdd(&accum[0], s);
    atomicMax((unsigned int*)&accum[1], __float_as_uint(mx));  // mx >= 0
    atomicAdd(&accum[2], cc);
  }
}

// ---------------- finalize scalars ------------------------------------------
__global__ void finalize_kernel(const float* __restrict__ accum,
                                float* __restrict__ out, int n_con, int out_size) {
  if (blockIdx.x == 0 && threadIdx.x == 0) {
    float mean = accum[0] / (float)n_con;
    float mx   = __uint_as_float(*(const unsigned int*)&accum[1]);
    if (out_size > 0) out[0] = LAMBDA_MEAN * mean + LAMBDA_MAX * mx;
    if (out_size > 1) out[1] = mean;
    if (out_size > 2) out[2] = mx;
    if (out_size > 3) out[3] = accum[2];
  }
}

extern "C" void kernel_launch(void* const* d_in, const int* in_sizes, int n_in,
                              void* d_out, int out_size, void* d_ws, size_t ws_size,
                              hipStream_t stream) {
  const float*     prob_bin = (const float*)d_in[0];
  const float*     logits   = (const float*)d_in[1];
  const float*     offs     = (const float*)d_in[2];
  const float*     pred_lg  = (const float*)d_in[3];
  const float*     ef       = (const float*)d_in[4];
  const float*     cf       = (const float*)d_in[5];
  const float*     vf       = (const float*)d_in[6];
  const long long* idx_bin  = (const long long*)d_in[7];
  const long long* idx_sm   = (const long long*)d_in[8];
  const long long* idx_lg   = (const long long*)d_in[9];
  const long long* vt       = (const long long*)d_in[10];
  const long long* eidx     = (const long long*)d_in[11];

  const int NB   = in_sizes[0];
  const int NS   = in_sizes[2];
  const int NL   = in_sizes[3];
  const int E    = in_sizes[4];
  const int NCON = in_sizes[5] / 8;
  const int NV   = in_sizes[6] / 16;

  float* x     = (float*)d_ws;      // NV floats
  float* Ax    = x + NV;            // NCON floats
  float* accum = Ax + NCON;         // 4 floats

  const int B = 256;
  const int initN = (NV > NCON) ? NV : NCON;

  init_ws_kernel<<<(initN + B - 1) / B, B, 0, stream>>>(x, NV, Ax, NCON, accum);
  scatter_kernel<<<(NB + B - 1) / B, B, 0, stream>>>(x, idx_bin, prob_bin, NB);
  {
    const int ntiles = (NS + 15) / 16;          // one wave per 16-row tile
    const int wavesPerBlock = B / 32;           // 8
    const int blocks = (ntiles + wavesPerBlock - 1) / wavesPerBlock;
    softmax_ev_wmma_kernel<<<blocks, B, 0, stream>>>(x, logits, offs, idx_sm, NS);
  }
  scatter_kernel<<<(NL + B - 1) / B, B, 0, stream>>>(x, idx_lg, pred_lg, NL);
  apply_lp_kernel<<<(NV + B - 1) / B, B, 0, stream>>>(x, vf, vt, NV);
  spmv_kernel<<<(E + B - 1) / B, B, 0, stream>>>(eidx, ef, x, Ax, E);
  reduce_kernel<<<(NCON + B - 1) / B, B, 0, stream>>>(Ax, cf, accum, NCON);
  finalize_kernel<<<1, 32, 0, stream>>>(accum, (float*)d_out, NCON, out_size);
}